// ReMaxKv_20117626814808
// MI455X (gfx1250) — compile-verified
//
#include <hip/hip_runtime.h>
#include <hip/hip_bf16.h>

#define AS1 __attribute__((address_space(1)))
#define AS3 __attribute__((address_space(3)))

typedef int      v4i   __attribute__((ext_vector_type(4)));
typedef unsigned u32x4 __attribute__((ext_vector_type(4)));
typedef int      i32x4 __attribute__((ext_vector_type(4)));
typedef int      i32x8 __attribute__((ext_vector_type(8)));

#define NCOLS    32768
#define NTHREADS 512
#define NWAVES   (NTHREADS / 32)
#define TOPK     64

// Monotone key: larger float -> larger unsigned key.
__device__ __forceinline__ unsigned key_of(float x) {
  unsigned u = __float_as_uint(x);
  return (u & 0x80000000u) ? ~u : (u | 0x80000000u);
}
__device__ __forceinline__ float val_of_key(unsigned k) {
  unsigned u = (k & 0x80000000u) ? (k ^ 0x80000000u) : ~k;
  return __uint_as_float(u);
}

__global__ __launch_bounds__(NTHREADS, 1) void remax_topk_kernel(
    const float* __restrict__ x, float* __restrict__ out) {
  // Row cache: 128 KB of the WGP's 320 KB LDS. Plus radix-select scratch.
  __shared__ __align__(16) float row[NCOLS];          // 131072 B
  __shared__ unsigned hist[NWAVES][256];              //  16384 B (per-wave privatized)
  __shared__ unsigned total[256];                     //   1024 B
  __shared__ float    wsum_relu[NWAVES];
  __shared__ float    wsum_top[NWAVES];
  __shared__ unsigned wsum_cnt[NWAVES];
  __shared__ unsigned sh_prefix;
  __shared__ unsigned sh_krem;
  __shared__ float    sh_mag;
  __shared__ float    sh_magk;

  const int tid  = threadIdx.x;
  const int wave = tid >> 5;
  const int lane = tid & 31;
  const size_t rowbase = (size_t)blockIdx.x * (size_t)NCOLS;
  const float* gx = x + rowbase;

  // ---- Stage 1: global row -> LDS ----
#if defined(__gfx1250__) && __has_builtin(__builtin_amdgcn_tensor_load_to_lds)
  // Tensor Data Mover: one DMA descriptor moves the whole 128 KB row.
  if (wave == 0) {
    const unsigned lds_off = (unsigned)(size_t)(AS3 char*)(char*)row;
    const unsigned long long ga = (unsigned long long)(size_t)gx;

    // D# group 0: count=1 | lds_addr | global_addr[56:0] | type=2
    u32x4 g0;
    g0[0] = 0x1u;                                            // count=1, is_restore=0
    g0[1] = lds_off;                                         // lds_addr
    g0[2] = (unsigned)(ga & 0xFFFFFFFFu);                    // global_addr[31:0]
    g0[3] = (unsigned)((ga >> 32) & 0x01FFFFFFu) | (2u << 30); // addr[56:32], type=2

    // D# group 1: data_size=4B, tensor_dim0=32768, tensor_dim1=1,
    //             tile_dim0=32768, tile_dim1=1, strides=32768
    i32x8 g1;
    g1[0] = 0x00020000;        // workgroup_mask=0, data_size=2 (4 bytes)
    g1[1] = (int)0x80000000u;  // bits[47:32]=0, tensor_dim0[15:0]=0x8000 << 16
    g1[2] = 0x00010000;        // tensor_dim0[31:16]=0, tensor_dim1[15:0]=1 << 16
    g1[3] = (int)0x80000000u;  // tensor_dim1[31:16]=0, tile_dim0=0x8000 << 16
    g1[4] = 0x00000001;        // tile_dim1=1, tile_dim2=0 (unused)
    g1[5] = 32768;             // tensor_dim0_stride[31:0]
    g1[6] = (int)0x80000000u;  // stride0[47:32]=0, tensor_dim1_stride[15:0]=0x8000 << 16
    g1[7] = 0;                 // tensor_dim1_stride[47:16]=0

    // D# group 2: tensor_dim2=1, tensor_dim3=1, dim2_stride=32768, tile_dim3=0
    i32x4 g2;
    g2[0] = 1;
    g2[1] = 1;
    g2[2] = 32768;
    g2[3] = 0;

    // D# group 3: dim3_stride=32768, tensor_dim4=1, tile_dim4=0
    i32x4 g3;
    g3[0] = 32768;
    g3[1] = 0x00010000;
    g3[2] = 0;
    g3[3] = 0;

    // VADDR4 slot: unused per ISA (set to NULL/zeros).
    i32x8 g4 = {0, 0, 0, 0, 0, 0, 0, 0};

    __builtin_amdgcn_tensor_load_to_lds(g0, g1, g2, g3, g4, 0);
#if __has_builtin(__builtin_amdgcn_s_wait_tensorcnt)
    __builtin_amdgcn_s_wait_tensorcnt(0);
#else
    asm volatile("s_wait_tensorcnt 0" ::: "memory");
#endif
  }
#elif defined(__gfx1250__) && __has_builtin(__builtin_amdgcn_global_load_async_to_lds_b128)
  {
    const v4i* gv = (const v4i*)gx;
    v4i* lv = (v4i*)row;
    #pragma unroll 4
    for (int i = tid; i < NCOLS / 4; i += NTHREADS) {
      __builtin_amdgcn_global_load_async_to_lds_b128(
          (AS1 v4i*)(v4i*)(gv + i), (AS3 v4i*)(lv + i), 0, 0);
    }
#if __has_builtin(__builtin_amdgcn_s_wait_asynccnt)
    __builtin_amdgcn_s_wait_asynccnt(0);
#else
    asm volatile("s_wait_asynccnt 0" ::: "memory");
#endif
  }
#else
  for (int i = tid; i < NCOLS / 4; i += NTHREADS) {
    ((float4*)row)[i] = ((const float4*)gx)[i];
  }
#endif
  __syncthreads();

  // ---- Stage 2: radix select the key of the 64th-largest element ----
  unsigned prefix = 0;  // known high bits of the pivot key (low bits zero)
  unsigned krem   = TOPK;
  unsigned* hflat = &hist[0][0];

  #pragma unroll
  for (int round = 0; round < 4; ++round) {
    const int shift = 24 - 8 * round;
    for (int i = tid; i < NWAVES * 256; i += NTHREADS) hflat[i] = 0;
    __syncthreads();

    const unsigned hi_mask = (round == 0) ? 0u : (0xFFFFFFFFu << (shift + 8));
    for (int i = tid; i < NCOLS; i += NTHREADS) {
      unsigned k = key_of(row[i]);
      if ((k & hi_mask) == prefix) {
        atomicAdd(&hist[wave][(k >> shift) & 0xFFu], 1u);
      }
    }
    __syncthreads();

    if (tid < 256) {
      unsigned s = 0;
      #pragma unroll
      for (int w = 0; w < NWAVES; ++w) s += hist[w][tid];
      total[tid] = s;
    }
    __syncthreads();

    if (tid == 0) {
      unsigned cum = 0;
      int b = 255;
      for (; b > 0; --b) {
        unsigned c = total[b];
        if (cum + c >= krem) break;
        cum += c;
      }
      sh_prefix = prefix | ((unsigned)b << shift);
      sh_krem   = krem - cum;
    }
    __syncthreads();
    prefix = sh_prefix;
    krem   = sh_krem;
    __syncthreads();
  }
  const unsigned pivot_key = prefix;
  const float    pivot_val = val_of_key(pivot_key);

  // ---- Stage 3: one LDS sweep for relu-sum, >pivot sum, >pivot count ----
  float    s_relu = 0.0f, s_top = 0.0f;
  unsigned c_gt = 0;
  for (int i = tid; i < NCOLS; i += NTHREADS) {
    float v = row[i];
    s_relu += fmaxf(v, 0.0f);
    if (key_of(v) > pivot_key) { s_top += v; c_gt += 1u; }
  }
  #pragma unroll
  for (int off = 16; off > 0; off >>= 1) {
    s_relu += __shfl_down(s_relu, off, 32);
    s_top  += __shfl_down(s_top,  off, 32);
    c_gt   += __shfl_down(c_gt,   off, 32);
  }
  if (lane == 0) { wsum_relu[wave] = s_relu; wsum_top[wave] = s_top; wsum_cnt[wave] = c_gt; }
  __syncthreads();
  if (tid == 0) {
    float a = 0.0f, b = 0.0f;
    unsigned c = 0;
    #pragma unroll
    for (int w = 0; w < NWAVES; ++w) { a += wsum_relu[w]; b += wsum_top[w]; c += wsum_cnt[w]; }
    sh_mag  = a;
    sh_magk = b + (float)((int)TOPK - (int)c) * pivot_val;  // tie handling at pivot
  }
  __syncthreads();

  const float factor = sh_magk / sh_mag;  // may be inf/nan if mag==0; guarded below

  // ---- Stage 4: elementwise scale and store (b128 stores) ----
  float* gout = out + rowbase;
  for (int i = tid; i < NCOLS / 4; i += NTHREADS) {
    float4 v = ((const float4*)row)[i];
    float4 o;
    o.x = fmaxf(v.x, 0.0f) * factor;
    o.y = fmaxf(v.y, 0.0f) * factor;
    o.z = fmaxf(v.z, 0.0f) * factor;
    o.w = fmaxf(v.w, 0.0f) * factor;
    if (__isnanf(o.x) || __isinff(o.x)) o.x = 0.0f;
    if (__isnanf(o.y) || __isinff(o.y)) o.y = 0.0f;
    if (__isnanf(o.z) || __isinff(o.z)) o.z = 0.0f;
    if (__isnanf(o.w) || __isinff(o.w)) o.w = 0.0f;
    ((float4*)gout)[i] = o;
  }
}

extern "C" void kernel_launch(void* const* d_in, const int* in_sizes, int n_in,
                              void* d_out, int out_size, void* d_ws, size_t ws_size,
                              hipStream_t stream) {
  const float* x = (const float*)d_in[0];
  float* out = (float*)d_out;
  int rows = 4096;
  if (n_in > 0 && in_sizes && in_sizes[0] > 0) rows = in_sizes[0] / NCOLS;
  remax_topk_kernel<<<rows, NTHREADS, 0, stream>>>(x, out);
  (void)d_ws; (void)ws_size; (void)out_size;
}